// GraphSAGEBackbone_26731876451057
// MI455X (gfx1250) — compile-verified
//
#include <hip/hip_runtime.h>

#define N_NODES 100000
#define N_EDGES 1600000
#define D 64
#define M_TILES ((N_NODES + 15) / 16)   // 6250 (exact: 100000 = 6250*16)
#define N_TILES (D / 16)                // 4

typedef float v2f __attribute__((ext_vector_type(2)));
typedef float v8f __attribute__((ext_vector_type(8)));

// ---------------- utility kernels ----------------

__global__ void zero_f32(float* __restrict__ p, int n) {
    int i = blockIdx.x * blockDim.x + threadIdx.x;
    if (i < n) p[i] = 0.0f;
}

__global__ void zero_f128(float4* __restrict__ p, int n4) {
    int i = blockIdx.x * blockDim.x + threadIdx.x;
    if (i < n4) p[i] = make_float4(0.f, 0.f, 0.f, 0.f);
}

__global__ void count_deg(const int* __restrict__ dst, float* __restrict__ deg, int ne) {
    int e = blockIdx.x * blockDim.x + threadIdx.x;
    if (e < ne) atomicAdd(&deg[dst[e]], 1.0f);
}

__global__ void finalize_invdeg(float* __restrict__ deg, int n) {
    int i = blockIdx.x * blockDim.x + threadIdx.x;
    if (i < n) deg[i] = 1.0f / fmaxf(deg[i], 1.0f);
}

// Edge scatter: one thread per (edge, 4-float chunk). Gather float4 from the
// source row, scatter-add into agg[dst]. agg (25.6MB) is L2-resident on MI455X
// (192MB L2) so these fp32 atomics resolve in L2.
__global__ void scatter_add(const float* __restrict__ xin,
                            const int* __restrict__ src,
                            const int* __restrict__ dst,
                            float* __restrict__ agg, int ne) {
    int t = blockIdx.x * blockDim.x + threadIdx.x;      // up to 25.6M, fits int32
    if (t >= ne * 16) return;
    int e = t >> 4;
    int c = (t & 15) << 2;                              // feature chunk offset
    int s = src[e];
    int d = dst[e];
    const float4 v = *(const float4*)(xin + (long long)s * D + c);
    float* ap = agg + (long long)d * D + c;
    atomicAdd(ap + 0, v.x);
    atomicAdd(ap + 1, v.y);
    atomicAdd(ap + 2, v.z);
    atomicAdd(ap + 3, v.w);
}

// Fused SAGE dense stage: out = (agg * invdeg) @ Wl + xin @ Wr + b, optional ReLU.
// One wave32 computes one 16x16 output tile via V_WMMA_F32_16X16X4_F32,
// 16 k-steps x 2 weight matrices = 32 WMMA ops, fully unrolled.
//
// A-fragment (16x4 f32): lane&15 = row M; VGPR v, half h=(lane>>4): K = h*2 + v.
// B-fragment (4x16 f32): lane&15 = col N; VGPR v, half h: K = h*2 + v.
// C/D (16x16 f32, 8 VGPRs): VGPR r, half h: M = r + h*8; N = lane&15.
__global__ __launch_bounds__(256)
void sage_gemm(const float* __restrict__ agg, const float* __restrict__ invdeg,
               const float* __restrict__ xin,
               const float* __restrict__ Wl, const float* __restrict__ Wr,
               const float* __restrict__ bias, float* __restrict__ out, int relu) {
    const int lane = threadIdx.x & 31;
    const int wid  = blockIdx.x * (blockDim.x >> 5) + (threadIdx.x >> 5);
    const int mt = wid / N_TILES;
    const int nt = wid % N_TILES;
    if (mt >= M_TILES) return;                  // wave-uniform guard (EXEC stays all-1s)

    const int half = lane >> 4;
    const int l15  = lane & 15;
    const int mrow = mt * 16 + l15;             // A-row held by this lane
    const int ncol = nt * 16 + l15;             // B/C column held by this lane

    const float idg = invdeg[mrow];             // fold mean-normalization into A load
    const float* aggRow = agg + (long long)mrow * D;
    const float* xRow   = xin + (long long)mrow * D;

    v8f c = {0.f, 0.f, 0.f, 0.f, 0.f, 0.f, 0.f, 0.f};

#pragma unroll
    for (int kk = 0; kk < D / 4; ++kk) {
        const int k = kk * 4 + half * 2;        // K indices {k, k+1} for this lane
        v2f aL, aR, bL, bR;
        float2 t;
        t = *(const float2*)(aggRow + k); aL[0] = t.x * idg; aL[1] = t.y * idg;
        t = *(const float2*)(xRow + k);   aR[0] = t.x;       aR[1] = t.y;
        bL[0] = Wl[(k + 0) * D + ncol];   bL[1] = Wl[(k + 1) * D + ncol];
        bR[0] = Wr[(k + 0) * D + ncol];   bR[1] = Wr[(k + 1) * D + ncol];
        c = __builtin_amdgcn_wmma_f32_16x16x4_f32(false, aL, false, bL, (short)0, c, false, false);
        c = __builtin_amdgcn_wmma_f32_16x16x4_f32(false, aR, false, bR, (short)0, c, false, false);
    }

    const float bn = bias[ncol];
#pragma unroll
    for (int r = 0; r < 8; ++r) {
        int m = mt * 16 + half * 8 + r;
        float v = c[r] + bn;
        if (relu) v = fmaxf(v, 0.0f);
        out[(long long)m * D + ncol] = v;
    }
}

// ---------------- launch ----------------

static void run_layer(const float* xin, const int* src, const int* dst,
                      const float* Wl, const float* Wr, const float* b,
                      float* agg, const float* invdeg, float* out, int relu,
                      hipStream_t stream) {
    const int n4 = N_NODES * (D / 4);                        // 1.6M float4s
    zero_f128<<<(n4 + 255) / 256, 256, 0, stream>>>((float4*)agg, n4);
    const int st = N_EDGES * 16;                             // 25.6M threads
    scatter_add<<<(st + 255) / 256, 256, 0, stream>>>(xin, src, dst, agg, N_EDGES);
    const int tiles = M_TILES * N_TILES;                     // 25000 waves
    sage_gemm<<<(tiles + 7) / 8, 256, 0, stream>>>(agg, invdeg, xin, Wl, Wr, b, out, relu);
}

extern "C" void kernel_launch(void* const* d_in, const int* in_sizes, int n_in,
                              void* d_out, int out_size, void* d_ws, size_t ws_size,
                              hipStream_t stream) {
    const float* x   = (const float*)d_in[0];
    const int*   ei  = (const int*)d_in[1];      // [2, N_EDGES] row-major, int32
    const int*   src = ei;
    const int*   dst = ei + N_EDGES;
    const float* Wl0 = (const float*)d_in[2];
    const float* Wr0 = (const float*)d_in[3];
    const float* b0  = (const float*)d_in[4];
    const float* Wl1 = (const float*)d_in[5];
    const float* Wr1 = (const float*)d_in[6];
    const float* b1  = (const float*)d_in[7];
    const float* Wl2 = (const float*)d_in[8];
    const float* Wr2 = (const float*)d_in[9];
    const float* b2  = (const float*)d_in[10];

    const size_t FEAT_BYTES = (size_t)N_NODES * D * sizeof(float);   // 25.6 MB
    char* ws = (char*)d_ws;
    float* agg    = (float*)(ws);
    float* h0     = (float*)(ws + FEAT_BYTES);
    float* h1     = (float*)(ws + 2 * FEAT_BYTES);
    float* invdeg = (float*)(ws + 3 * FEAT_BYTES);

    // Degree -> 1/max(deg,1), computed once and reused by all three layers.
    zero_f32<<<(N_NODES + 255) / 256, 256, 0, stream>>>(invdeg, N_NODES);
    count_deg<<<(N_EDGES + 255) / 256, 256, 0, stream>>>(dst, invdeg, N_EDGES);
    finalize_invdeg<<<(N_NODES + 255) / 256, 256, 0, stream>>>(invdeg, N_NODES);

    // Layer 0: x -> h0 (ReLU)
    run_layer(x,  src, dst, Wl0, Wr0, b0, agg, invdeg, h0, 1, stream);
    // Layer 1: h0 -> h1 (ReLU)
    run_layer(h0, src, dst, Wl1, Wr1, b1, agg, invdeg, h1, 1, stream);
    // Layer 2: h1 -> out (no ReLU)
    run_layer(h1, src, dst, Wl2, Wr2, b2, agg, invdeg, (float*)d_out, 0, stream);
}